// SAGPool_214748365117
// MI455X (gfx1250) — compile-verified
//
#include <hip/hip_runtime.h>
#include <math.h>

// ---------------- problem constants (from reference) ----------------
#define N_GRAPHS 100
#define N_PER    1000
#define N_NODES  (N_GRAPHS * N_PER)      // 100000
#define E_EDGES  3200000
#define C_CH     256
#define K_KEEP   800
#define N_OUT    (N_GRAPHS * K_KEEP)     // 80000

typedef __attribute__((ext_vector_type(2))) float v2f;
typedef __attribute__((ext_vector_type(8))) float v8f;

// ---------------- kernel 0: init scratch ----------------
__global__ __launch_bounds__(256) void init_kernel(float* __restrict__ deg,
                                                   float* __restrict__ agg,
                                                   int* __restrict__ mask,
                                                   int* __restrict__ newpos) {
  int n = blockIdx.x * blockDim.x + threadIdx.x;
  if (n >= N_NODES) return;
  deg[n] = 1.0f;     // +1.0 self-loop term of reference degree
  agg[n] = 0.0f;
  mask[n] = 0;
  newpos[n] = -1;
}

// ---------------- kernel 1: h = x @ w via V_WMMA_F32_16X16X4_F32 ----------------
// Each wave computes 16 node scores. A = 16x4 fp32 tile of x, B = 4x16 with the
// weight chunk in column 0 (all other columns zero). Accumulate D over K=256 in
// 64 chunks of 4. D column 0 lives in lane 0 (M=0..7) and lane 16 (M=8..15).
__global__ __launch_bounds__(256) void matvec_wmma_kernel(const float* __restrict__ x,
                                                          const float* __restrict__ w,
                                                          float* __restrict__ h) {
  const int lane = threadIdx.x & 31;
  const int wave = threadIdx.x >> 5;
  const int m    = lane & 15;   // matrix row for A, column for B
  const int hi   = lane >> 4;   // 0 -> K pair {0,1}; 1 -> K pair {2,3}
  const int base = (blockIdx.x * 8 + wave) * 16;
  if (base >= N_NODES) return;  // wave-uniform: EXEC stays all-ones for WMMA

  const float* xrow = x + (size_t)(base + m) * C_CH + 2 * hi;
  const float* wrow = w + 2 * hi;

  v8f acc = {};
  #pragma unroll 4
  for (int k0 = 0; k0 < C_CH; k0 += 4) {
    v2f a = *(const v2f*)(xrow + k0);         // A: x[node][k0+2*hi .. +1]
    v2f wv = *(const v2f*)(wrow + k0);        // broadcast load of weight chunk
    v2f b;
    b.x = (m == 0) ? wv.x : 0.0f;             // B column 0 carries the weight
    b.y = (m == 0) ? wv.y : 0.0f;
    acc = __builtin_amdgcn_wmma_f32_16x16x4_f32(
        /*neg_a=*/false, a, /*neg_b=*/false, b,
        /*c_mod=*/(short)0, acc, /*reuse_a=*/false, /*reuse_b=*/false);
  }

  if (m == 0) {                               // lane 0: M=0..7, lane 16: M=8..15
    float* out = h + base + 8 * hi;
    out[0] = acc[0]; out[1] = acc[1]; out[2] = acc[2]; out[3] = acc[3];
    out[4] = acc[4]; out[5] = acc[5]; out[6] = acc[6]; out[7] = acc[7];
  }
}

// ---------------- kernel 2: degree count (L2-resident atomics) ----------------
__global__ __launch_bounds__(256) void degree_kernel(const int* __restrict__ ei,
                                                     float* __restrict__ deg) {
  int e = blockIdx.x * blockDim.x + threadIdx.x;
  if (e >= E_EDGES) return;
  atomicAdd(&deg[ei[E_EDGES + e]], 1.0f);     // col endpoint
}

// ---------------- kernel 3: deg -> rsqrt(deg) in place ----------------
__global__ __launch_bounds__(256) void rsqrt_kernel(float* __restrict__ deg) {
  int n = blockIdx.x * blockDim.x + threadIdx.x;
  if (n >= N_NODES) return;
  deg[n] = rsqrtf(deg[n]);
}

// ---------------- kernel 4: agg[col] += dinv[row]*dinv[col]*h[row] ----------------
__global__ __launch_bounds__(256) void aggregate_kernel(const int* __restrict__ ei,
                                                        const float* __restrict__ dinv,
                                                        const float* __restrict__ h,
                                                        float* __restrict__ agg) {
  int e = blockIdx.x * blockDim.x + threadIdx.x;
  if (e >= E_EDGES) return;
  int r = ei[e];
  int c = ei[E_EDGES + e];
  atomicAdd(&agg[c], dinv[r] * dinv[c] * h[r]);
}

// ---------------- kernel 5: score = agg + dinv^2 * h + bias ----------------
__global__ __launch_bounds__(256) void score_kernel(const float* __restrict__ agg,
                                                    const float* __restrict__ dinv,
                                                    const float* __restrict__ h,
                                                    const float* __restrict__ bias,
                                                    float* __restrict__ score) {
  int n = blockIdx.x * blockDim.x + threadIdx.x;
  if (n >= N_NODES) return;
  float di = dinv[n];
  score[n] = agg[n] + di * di * h[n] + bias[0];
}

// ---------------- kernel 6: per-graph sorted top-K (bitonic, 1024 in LDS) ----------------
__global__ __launch_bounds__(1024) void topk_kernel(const float* __restrict__ score,
                                                    int* __restrict__ perm_out,
                                                    int* __restrict__ batch_out,
                                                    int* __restrict__ mask,
                                                    int* __restrict__ newpos) {
  __shared__ float sk[1024];
  __shared__ int   si[1024];
  const int g = blockIdx.x;
  const int t = threadIdx.x;

  sk[t] = (t < N_PER) ? score[g * N_PER + t] : -__builtin_huge_valf();
  si[t] = t;
  __syncthreads();

  // bitonic sort: descending by score, ties broken by ascending index (top_k semantics)
  for (int k = 2; k <= 1024; k <<= 1) {
    for (int j = k >> 1; j > 0; j >>= 1) {
      int p = t ^ j;
      if (p > t) {
        float s0 = sk[t], s1 = sk[p];
        int   i0 = si[t], i1 = si[p];
        bool firstWins = (s0 > s1) || (s0 == s1 && i0 < i1);
        bool desc = ((t & k) == 0);
        if (desc ? !firstWins : firstWins) {
          sk[t] = s1; sk[p] = s0; si[t] = i1; si[p] = i0;
        }
      }
      __syncthreads();
    }
  }

  if (t < K_KEEP) {
    int node = g * N_PER + si[t];
    int pos  = g * K_KEEP + t;
    perm_out[pos]  = node;
    batch_out[pos] = g;
    mask[node]   = 1;
    newpos[node] = pos;
  }
}

// ---------------- kernel 7: x_out = x[perm] * tanh(score[perm]) (float4 stream) ----------------
__global__ __launch_bounds__(256) void gather_scale_kernel(const float* __restrict__ x,
                                                           const float* __restrict__ score,
                                                           const int* __restrict__ perm,
                                                           float* __restrict__ xout) {
  int gid = blockIdx.x * blockDim.x + threadIdx.x;        // float4 index
  const int TOT4 = N_OUT * (C_CH / 4);                    // 5,120,000
  if (gid >= TOT4) return;
  int i  = gid >> 6;          // output row
  int c4 = gid & 63;          // float4 within row
  int node = perm[i];
  float tv = tanhf(score[node]);
  float4 v = ((const float4*)(x + (size_t)node * C_CH))[c4];
  float4 o = make_float4(v.x * tv, v.y * tv, v.z * tv, v.w * tv);
  ((float4*)(xout + (size_t)i * C_CH))[c4] = o;
}

// ---------------- kernel 8: filter_adj ----------------
__global__ __launch_bounds__(256) void filter_adj_kernel(const int* __restrict__ ei,
                                                         const float* __restrict__ ea,
                                                         const int* __restrict__ mask,
                                                         const int* __restrict__ newpos,
                                                         int* __restrict__ ei_new,
                                                         float* __restrict__ ea_new) {
  int e = blockIdx.x * blockDim.x + threadIdx.x;
  if (e >= E_EDGES) return;
  int r = ei[e];
  int c = ei[E_EDGES + e];
  bool keep = (mask[r] != 0) && (mask[c] != 0);
  ei_new[e]           = keep ? newpos[r] : -1;
  ei_new[E_EDGES + e] = keep ? newpos[c] : -1;
  ea_new[e]           = keep ? ea[e] : 0.0f;
}

// ---------------- launcher ----------------
extern "C" void kernel_launch(void* const* d_in, const int* in_sizes, int n_in,
                              void* d_out, int out_size, void* d_ws, size_t ws_size,
                              hipStream_t stream) {
  const float* x    = (const float*)d_in[0];   // [N, C]
  const int*   ei   = (const int*)d_in[1];     // [2, E] flat
  const float* ea   = (const float*)d_in[2];   // [E]
  // d_in[3] = batch (recomputable: node/1000)
  const float* w    = (const float*)d_in[4];   // [C]
  const float* bias = (const float*)d_in[5];   // [1]

  // workspace layout (floats/ints, 6*N*4 bytes = 2.4 MB)
  float* h      = (float*)d_ws;
  float* deg    = h + N_NODES;        // becomes dinv after rsqrt_kernel
  float* agg    = deg + N_NODES;
  float* score  = agg + N_NODES;
  int*   mask   = (int*)(score + N_NODES);
  int*   newpos = mask + N_NODES;

  // output layout: x_out | ei_new | ea_new | batch_out | perm
  float* xout      = (float*)d_out;
  int*   ei_new    = (int*)(xout + (size_t)N_OUT * C_CH);
  float* ea_new    = (float*)(ei_new + 2 * E_EDGES);
  int*   batch_out = (int*)(ea_new + E_EDGES);
  int*   perm_out  = batch_out + N_OUT;

  const int TB = 256;
  init_kernel<<<(N_NODES + TB - 1) / TB, TB, 0, stream>>>(deg, agg, mask, newpos);

  // 8 waves/block * 16 nodes/wave = 128 nodes per block
  matvec_wmma_kernel<<<(N_NODES + 127) / 128, TB, 0, stream>>>(x, w, h);

  degree_kernel<<<(E_EDGES + TB - 1) / TB, TB, 0, stream>>>(ei, deg);
  rsqrt_kernel<<<(N_NODES + TB - 1) / TB, TB, 0, stream>>>(deg);
  aggregate_kernel<<<(E_EDGES + TB - 1) / TB, TB, 0, stream>>>(ei, deg, h, agg);
  score_kernel<<<(N_NODES + TB - 1) / TB, TB, 0, stream>>>(agg, deg, h, bias, score);

  topk_kernel<<<N_GRAPHS, 1024, 0, stream>>>(score, perm_out, batch_out, mask, newpos);

  gather_scale_kernel<<<(N_OUT * (C_CH / 4) + TB - 1) / TB, TB, 0, stream>>>(x, score, perm_out, xout);
  filter_adj_kernel<<<(E_EDGES + TB - 1) / TB, TB, 0, stream>>>(ei, ea, mask, newpos, ei_new, ea_new);
}